// SparseInterpolate3D_51015621542409
// MI455X (gfx1250) — compile-verified
//
#include <hip/hip_runtime.h>
#include <stdint.h>

// Grid constants from the reference (spconv zyx convention, xyz voxel size)
#define ZDIM 41
#define YDIM 1440
#define XDIM 1440
#define CCH  64
#define VSX  0.075f
#define VSY  0.075f
#define VSZ  0.2f

#define EMPTY_SLOT 0xFFFFFFFFFFFFFFFFull

// ---------------------------------------------------------------------------
// Hash table: slot = (key << 32) | voxel_row.  key < 2^31 so slot != EMPTY.
// ---------------------------------------------------------------------------
__global__ void init_table_kernel(unsigned long long* __restrict__ table, unsigned n) {
  unsigned i = blockIdx.x * blockDim.x + threadIdx.x;
  if (i < n) table[i] = EMPTY_SLOT;
}

__global__ void build_table_kernel(const int* __restrict__ indices, int n,
                                   unsigned long long* __restrict__ table,
                                   unsigned mask) {
  int i = blockIdx.x * blockDim.x + threadIdx.x;
  if (i >= n) return;
  int b = indices[4 * i + 0], z = indices[4 * i + 1];
  int y = indices[4 * i + 2], x = indices[4 * i + 3];
  unsigned key = (unsigned)((((b * ZDIM + z) * YDIM + y) * XDIM) + x);
  unsigned long long packed = ((unsigned long long)key << 32) | (unsigned)i;
  unsigned h = (key * 2654435761u) & mask;
  for (;;) {
    unsigned long long prev = atomicCAS(&table[h], EMPTY_SLOT, packed);
    if (prev == EMPTY_SLOT) return;
    if ((unsigned)(prev >> 32) == key) {
      // duplicate voxel key: keep smallest original row (stable-argsort semantics)
      atomicMin(&table[h], packed);
      return;
    }
    h = (h + 1) & mask;
  }
}

// ---------------------------------------------------------------------------
// Per-lane probe for one group of 4 points: lane = p_local(2b)*8 + corner(3b).
// Produces trilinear weight and feature-row byte offset (0 / w=0 if absent).
// ---------------------------------------------------------------------------
__device__ __forceinline__ void probe_group(
    int g, int lane, int M, int nbatch, const float* __restrict__ xyz,
    const int* __restrict__ batch_cnt,
    const unsigned long long* __restrict__ table, unsigned mask, float& wgt,
    unsigned& voff) {
  const int p_local = lane >> 3;
  const int k = lane & 7;
  const int dx = k & 1, dy = (k >> 1) & 1, dz = (k >> 2) & 1;
  const int p = g * 4 + p_local;
  wgt = 0.0f;
  voff = 0;
  if (p < M) {
    float px = xyz[3 * p + 0], py = xyz[3 * p + 1], pz = xyz[3 * p + 2];
    int pb = 0, cum = 0;
    for (int bb = 0; bb < nbatch; ++bb) {
      cum += batch_cnt[bb];
      if (p >= cum) pb = bb + 1;
    }
    float ux = px / VSX - 0.5f, uy = py / VSY - 0.5f, uz = pz / VSZ - 0.5f;
    float fx0 = floorf(ux), fy0 = floorf(uy), fz0 = floorf(uz);
    int ix = (int)fx0 + dx, iy = (int)fy0 + dy, iz = (int)fz0 + dz;
    float fx = ux - fx0, fy = uy - fy0, fz = uz - fz0;
    float wx = dx ? fx : 1.0f - fx;
    float wy = dy ? fy : 1.0f - fy;
    float wz = dz ? fz : 1.0f - fz;
    bool inb = (ix >= 0) && (ix < XDIM) && (iy >= 0) && (iy < YDIM) &&
               (iz >= 0) && (iz < ZDIM);
    if (inb) {
      unsigned key = (unsigned)((((pb * ZDIM + iz) * YDIM + iy) * XDIM) + ix);
      unsigned h = (key * 2654435761u) & mask;
      int r = -1;
      for (;;) {
        unsigned long long v = table[h];
        if (v == EMPTY_SLOT) break;
        if ((unsigned)(v >> 32) == key) { r = (int)(v & 0xffffffffu); break; }
        h = (h + 1) & mask;
      }
      if (r >= 0) {
        wgt = wx * wy * wz;
        voff = (unsigned)r * (CCH * 4);  // <= 25.6MB, fits u32
      }
    }
  }
}

// ---------------------------------------------------------------------------
// Issue 16 async B128 gathers covering 32 feature rows (256B each) into one
// LDS slab: per issue t, lanes 0-15 move row 2t, lanes 16-31 move row 2t+1
// (16B per lane).  GVS mode: mem = SGPR64(features) + VGPR32(offset).
// Only 16 ASYNCcnt increments per group -> 32 in flight when double-buffered,
// safely inside the 6-bit counter.
// ---------------------------------------------------------------------------
__device__ __forceinline__ void issue_async_rows(unsigned slabBase, int lane,
                                                 unsigned voff,
                                                 const float* __restrict__ features) {
  const int hi = lane >> 4;                           // half-wave id
  const unsigned laneByte = (unsigned)(lane & 15) * 16u;
#pragma unroll
  for (int t = 0; t < 16; ++t) {
    const int row = 2 * t + hi;
    unsigned rowoff = (unsigned)__shfl((int)voff, row, 32);
    unsigned dst = slabBase + (unsigned)row * (CCH * 4) + laneByte;
    unsigned src = rowoff + laneByte;
    asm volatile("global_load_async_to_lds_b128 %0, %1, %2"
                 :
                 : "v"(dst), "v"(src), "s"(features)
                 : "memory");
  }
}

// ---------------------------------------------------------------------------
// Persistent, double-buffered interpolation kernel. One wave32 handles groups
// of 4 points; while slab[buf] is consumed, slab[buf^1] is being filled by the
// async engine (older batch drained with partial wait s_wait_asynccnt 0x10).
// ---------------------------------------------------------------------------
__global__ void __launch_bounds__(128)
interp_kernel(const float* __restrict__ features,
              const float* __restrict__ xyz,
              const int* __restrict__ batch_cnt, int nbatch,
              const unsigned long long* __restrict__ table, unsigned mask,
              float* __restrict__ out, int M, int G) {
  __shared__ __align__(16) float lbuf[2 * 4 * 32 * CCH];  // 2 bufs*4 waves*8KB

  const int tid = threadIdx.x;
  const int lane = tid & 31;
  const int waveInBlk = tid >> 5;
  const int waveId = blockIdx.x * (blockDim.x >> 5) + waveInBlk;
  const int totalWaves = gridDim.x * (blockDim.x >> 5);
  const unsigned ldsBase0 = (unsigned)(uintptr_t)(&lbuf[0]);
  const int hi = lane >> 4;   // half-wave: points {2*hi, 2*hi+1}
  const int cl = lane & 15;   // channel-lane: channels 4*cl .. 4*cl+3

  int g = waveId;
  if (g >= G) return;

  float wgt;
  unsigned voff;
  probe_group(g, lane, M, nbatch, xyz, batch_cnt, table, mask, wgt, voff);
  int buf = 0;
  issue_async_rows(ldsBase0 + (unsigned)((buf * 4 + waveInBlk) * 32 * CCH * 4),
                   lane, voff, features);

  for (;;) {
    const int gn = g + totalWaves;
    const bool nextValid = gn < G;  // wave-uniform
    float wgtN = 0.0f;
    unsigned voffN = 0;
    if (nextValid) {
      // overlap next group's probes + async issue with in-flight gathers
      probe_group(gn, lane, M, nbatch, xyz, batch_cnt, table, mask, wgtN,
                  voffN);
      issue_async_rows(
          ldsBase0 + (unsigned)(((buf ^ 1) * 4 + waveInBlk) * 32 * CCH * 4),
          lane, voffN, features);
      // in-order ASYNCcnt: <=16 outstanding means the older 16 have landed
      asm volatile("s_wait_asynccnt 0x10" ::: "memory");
    } else {
      asm volatile("s_wait_asynccnt 0x0" ::: "memory");
    }

    // ---- consume slab[buf]: half-wave owns 2 points, lane owns 4 channels --
    const float* wbuf = &lbuf[(buf * 4 + waveInBlk) * 32 * CCH];
#pragma unroll
    for (int pp = 0; pp < 2; ++pp) {
      const int p = hi * 2 + pp;  // point within group for this half-wave
      float a0 = 0.f, a1 = 0.f, a2 = 0.f, a3 = 0.f, ws = 0.f;
#pragma unroll
      for (int kk = 0; kk < 8; ++kk) {
        float wj = __shfl(wgt, p * 8 + kk, 32);
        const float4 v = *(const float4*)&wbuf[(p * 8 + kk) * CCH + cl * 4];
        a0 += wj * v.x;
        a1 += wj * v.y;
        a2 += wj * v.z;
        a3 += wj * v.w;
        ws += wj;
      }
      const int pg = g * 4 + p;
      if (pg < M) {
        float m = fmaxf(ws, 1e-8f);
        float4 o;
        o.x = a0 / m;
        o.y = a1 / m;
        o.z = a2 / m;
        o.w = a3 / m;
        *(float4*)&out[(size_t)pg * CCH + cl * 4] = o;
      }
    }

    if (!nextValid) break;
    g = gn;
    wgt = wgtN;
    voff = voffN;
    buf ^= 1;
  }
}

// ---------------------------------------------------------------------------
extern "C" void kernel_launch(void* const* d_in, const int* in_sizes, int n_in,
                              void* d_out, int out_size, void* d_ws,
                              size_t ws_size, hipStream_t stream) {
  const float* features  = (const float*)d_in[0];
  const int*   indices   = (const int*)d_in[1];
  const float* xyz       = (const float*)d_in[2];
  const int*   batch_cnt = (const int*)d_in[3];
  float*       out       = (float*)d_out;

  const int N      = in_sizes[1] / 4;  // voxels
  const int M      = in_sizes[2] / 3;  // points
  const int nbatch = in_sizes[3];

  // Hash table in workspace: prefer 2^18 slots (2MB, load 0.38); shrink to fit
  // ws_size but never below 2^17 (must stay > N so probes terminate).
  unsigned tableSize = 1u << 18;
  while ((unsigned long long)tableSize * 8ull > (unsigned long long)ws_size &&
         tableSize > (1u << 17))
    tableSize >>= 1;
  unsigned mask = tableSize - 1;
  unsigned long long* table = (unsigned long long*)d_ws;

  init_table_kernel<<<(tableSize + 255) / 256, 256, 0, stream>>>(table,
                                                                 tableSize);
  build_table_kernel<<<(N + 255) / 256, 256, 0, stream>>>(indices, N, table,
                                                          mask);

  const int G = (M + 3) / 4;              // point groups (4 pts per wave)
  int blocks = (G + 3) / 4;               // 4 waves per block
  if (blocks > 2048) blocks = 2048;       // persistent grid-stride above this
  if (blocks < 1) blocks = 1;
  interp_kernel<<<blocks, 128, 0, stream>>>(features, xyz, batch_cnt, nbatch,
                                            table, mask, out, M, G);
}